// window_l2_norm_28406913696463
// MI455X (gfx1250) — compile-verified
//
#include <hip/hip_runtime.h>
#include <math.h>

// Windowed L2 norm: out = sqrt(box11(reflect_pad((img1-img2)^2)) + 1e-8)
// NCHW (8,3,1024,1024) f32. Bandwidth-bound (~302MB min @ 23.3TB/s ~= 13us).
// Fused single pass: LDS-tiled separable box filter; vertical 11-tap sum done
// as a band-matrix multiply on the full-f32 matrix pipe (V_WMMA_F32_16X16X4_F32).

typedef __attribute__((ext_vector_type(2))) float v2f;
typedef __attribute__((ext_vector_type(8))) float v8f;

#define W_SIZE 11
#define PAD 5
#define IMG_H 1024
#define IMG_W 1024
#define TILE_H 32
#define TILE_W 64
#define SQ_H (TILE_H + 2 * PAD)   // 42 halo rows
#define SQ_W (TILE_W + 2 * PAD)   // 74 halo cols
#define SQ_STRIDE (SQ_W + 1)      // 75 (pad to dodge bank conflicts)
#define HS_H 48                   // 42 live rows + 6 zero rows for WMMA H2 reads
#define HS_STRIDE (TILE_W + 1)    // 65

__global__ __launch_bounds__(256) void window_l2_norm_28406913696463_kernel(
    const float* __restrict__ img1, const float* __restrict__ img2,
    float* __restrict__ out)
{
    __shared__ float sq[SQ_H * SQ_STRIDE];   // (a-b)^2 halo tile
    __shared__ float hs[HS_H * HS_STRIDE];   // horizontal 11-tap sums

    const int t = threadIdx.x;
    const int x0 = blockIdx.x * TILE_W;
    const int y0 = blockIdx.y * TILE_H;
    const size_t pbase = (size_t)blockIdx.z * (size_t)(IMG_H * IMG_W);
    const float* __restrict__ a = img1 + pbase;
    const float* __restrict__ b = img2 + pbase;
    float* __restrict__ o = out + pbase;

    // ---- Phase 1: load halo region, square the diff, reflect at borders ----
    for (int idx = t; idx < SQ_H * SQ_W; idx += 256) {
        int ly = idx / SQ_W;
        int lx = idx - ly * SQ_W;
        int gy = y0 - PAD + ly;
        int gx = x0 - PAD + lx;
        gy = (gy < 0) ? -gy : gy;
        gy = (gy > IMG_H - 1) ? (2 * (IMG_H - 1) - gy) : gy;
        gx = (gx < 0) ? -gx : gx;
        gx = (gx > IMG_W - 1) ? (2 * (IMG_W - 1) - gx) : gx;
        size_t gidx = (size_t)gy * IMG_W + gx;
        float d = a[gidx] - b[gidx];
        sq[ly * SQ_STRIDE + lx] = d * d;
    }
    // Zero hsum rows 42..47 (read by the M2 WMMA operand with 0 coefficients;
    // must be finite so 0*x stays 0).
    for (int i = t; i < (HS_H - SQ_H) * TILE_W; i += 256) {
        int r = SQ_H + i / TILE_W;
        int c = i - (i / TILE_W) * TILE_W;
        hs[r * HS_STRIDE + c] = 0.0f;
    }
    __syncthreads();

    // ---- Phase 2: horizontal 11-tap sliding-window sums (16 cols / thread) ----
    if (t < SQ_H * (TILE_W / 16)) {
        int r  = t % SQ_H;
        int xb = (t / SQ_H) * 16;
        const float* row = &sq[r * SQ_STRIDE + xb];
        float s = 0.0f;
        #pragma unroll
        for (int k = 0; k < W_SIZE; ++k) s += row[k];
        hs[r * HS_STRIDE + xb] = s;
        #pragma unroll
        for (int x = 1; x < 16; ++x) {
            s += row[x + W_SIZE - 1] - row[x - 1];
            hs[r * HS_STRIDE + xb + x] = s;
        }
    }
    __syncthreads();

    // ---- Phase 3: vertical 11-tap sum as band-matrix WMMA ----
    // out[i][n] = sum_{k=i..i+10} hs[rbase+k][n]
    //           = (M1 x H1)[i][n] + (M2 x H2)[i][n]
    // M1[i][k]  = 1 iff i <= k <= i+10          (k  in 0..15)
    // M2[i][k'] = 1 iff k' <= i-6               (k' in 0..9, chunk 3 all-zero)
    const int wave = t >> 5;          // 8 waves: 2 (rows) x 4 (cols) of 16x16
    const int lane = t & 31;
    const int wr = wave >> 2;
    const int wc = wave & 3;
    const int m    = lane & 15;       // A: row M / B: col N / D: col N
    const int half = lane >> 4;       // lanes 16-31 hold K+2 / rows M+8
    const int rbase = 16 * wr;
    const int cb    = 16 * wc;

    v8f acc = {};
    #pragma unroll
    for (int c = 0; c < 4; ++c) {     // M1 x H1, K = 0..15 in chunks of 4
        int k0 = 4 * c + 2 * half;
        int k1 = k0 + 1;
        v2f av, bv;
        av.x = (k0 >= m && k0 <= m + 10) ? 1.0f : 0.0f;
        av.y = (k1 >= m && k1 <= m + 10) ? 1.0f : 0.0f;
        bv.x = hs[(rbase + k0) * HS_STRIDE + cb + m];
        bv.y = hs[(rbase + k1) * HS_STRIDE + cb + m];
        acc = __builtin_amdgcn_wmma_f32_16x16x4_f32(
                  false, av, false, bv, (short)0, acc, false, false);
    }
    #pragma unroll
    for (int c = 0; c < 3; ++c) {     // M2 x H2, K' = 0..11 (>=10 is zero)
        int k0 = 4 * c + 2 * half;
        int k1 = k0 + 1;
        v2f av, bv;
        av.x = (k0 <= m - 6) ? 1.0f : 0.0f;
        av.y = (k1 <= m - 6) ? 1.0f : 0.0f;
        bv.x = hs[(rbase + 16 + k0) * HS_STRIDE + cb + m];
        bv.y = hs[(rbase + 16 + k1) * HS_STRIDE + cb + m];
        acc = __builtin_amdgcn_wmma_f32_16x16x4_f32(
                  false, av, false, bv, (short)0, acc, false, false);
    }

    // ---- Epilogue: sqrt(s + eps), scatter per documented C/D layout ----
    #pragma unroll
    for (int v = 0; v < 8; ++v) {
        int rowm = v + 8 * half;                  // VGPR v: M=v (lo half), M=v+8 (hi)
        int gy = y0 + rbase + rowm;
        int gx = x0 + cb + m;                     // N = lane & 15
        o[(size_t)gy * IMG_W + gx] = sqrtf(acc[v] + 1e-8f);
    }
}

extern "C" void kernel_launch(void* const* d_in, const int* in_sizes, int n_in,
                              void* d_out, int out_size, void* d_ws, size_t ws_size,
                              hipStream_t stream) {
    const float* img1 = (const float*)d_in[0];
    const float* img2 = (const float*)d_in[1];
    float* out = (float*)d_out;

    const int planes = in_sizes[0] / (IMG_H * IMG_W);   // B*C = 24
    dim3 grid(IMG_W / TILE_W, IMG_H / TILE_H, planes);  // 16 x 32 x 24
    dim3 block(256);
    hipLaunchKernelGGL(window_l2_norm_28406913696463_kernel,
                       grid, block, 0, stream, img1, img2, out);
}